// PrototypeLayer_71451075936309
// MI455X (gfx1250) — compile-verified
//
#include <hip/hip_runtime.h>
#include <hip/hip_bf16.h>

typedef float v2f __attribute__((ext_vector_type(2)));
typedef float v8f __attribute__((ext_vector_type(8)));

#define D_DIM 256

// ---------------------------------------------------------------------------
// Kernel 1: pnorm[p] = sum_k prototypes[p][k]^2
// ---------------------------------------------------------------------------
__global__ void pnorm_kernel(const float* __restrict__ protos,
                             float* __restrict__ pnorm, int P) {
    int p = blockIdx.x * blockDim.x + threadIdx.x;
    if (p >= P) return;
    const float4* row = reinterpret_cast<const float4*>(protos + (size_t)p * D_DIM);
    float s = 0.f;
    #pragma unroll 8
    for (int i = 0; i < D_DIM / 4; ++i) {
        float4 v = row[i];
        s += v.x * v.x + v.y * v.y + v.z * v.z + v.w * v.w;
    }
    pnorm[p] = s;
}

// ---------------------------------------------------------------------------
// Kernel 2: per wave: 16 rows of x. WMMA f32 16x16x4 distance matmul vs all
// P prototypes with register double-buffered fragments, argmin, gather,
// write proto + residual, per-wave loss partial.
// Block = 256 threads = 8 waves; each block covers 128 rows.
// ---------------------------------------------------------------------------
__global__ void vq_main(const float* __restrict__ x,
                        const float* __restrict__ protos,
                        const float* __restrict__ pnorm,
                        float* __restrict__ proto_out,
                        float* __restrict__ resid_out,
                        float* __restrict__ partials,
                        int P) {
    __shared__ int sIdx[8 * 16];

    const int lane = threadIdx.x & 31;
    const int wave = threadIdx.x >> 5;
    const int waveGlobal = blockIdx.x * 8 + wave;
    const int rowBase = waveGlobal * 16;

    const int m    = lane & 15;        // row within 16-row strip (A/B frag)
    const int koff = (lane >> 4) << 1; // K sub-offset 0 or 2

    // A fragment base pointer for this lane: x[rowBase + m][koff + k0 .. +1]
    const float* aptr = x + ((size_t)(rowBase + m)) * D_DIM + koff;

    float minv[8];
    int   mini[8];
    #pragma unroll
    for (int r = 0; r < 8; ++r) { minv[r] = 3.4e38f; mini[r] = 0; }

    // N-passes: 128 prototypes (8 tiles of 16) per pass
    for (int nBase = 0; nBase < P; nBase += 128) {
        // B fragment base: protos[nBase + nt*16 + m][koff + k0 .. +1]
        // nt handled via immediate offset (nt * 16 rows * 256 floats).
        const float* bptr = protos + ((size_t)(nBase + m)) * D_DIM + koff;

        // Prefetch next pass's B rows into L2/L0
        if (nBase + 128 < P) {
            #pragma unroll
            for (int nt = 0; nt < 8; ++nt)
                __builtin_prefetch(bptr + (size_t)(128 + nt * 16) * D_DIM, 0, 1);
        }

        v8f c[8];
        #pragma unroll
        for (int nt = 0; nt < 8; ++nt)
            c[nt] = (v8f){0.f,0.f,0.f,0.f,0.f,0.f,0.f,0.f};

        // ---- software-pipelined K loop: prefetch k+1 fragments into
        // registers while WMMAs for k execute ----
        v2f a;
        v2f b[8];
        a = *reinterpret_cast<const v2f*>(aptr);
        #pragma unroll
        for (int nt = 0; nt < 8; ++nt)
            b[nt] = *reinterpret_cast<const v2f*>(bptr + nt * 16 * D_DIM);

        #pragma unroll 2
        for (int k0 = 0; k0 < D_DIM; k0 += 4) {
            const int kn = (k0 + 4 < D_DIM) ? (k0 + 4) : k0;  // clamp at tail
            v2f an = *reinterpret_cast<const v2f*>(aptr + kn);
            v2f bn[8];
            #pragma unroll
            for (int nt = 0; nt < 8; ++nt)
                bn[nt] = *reinterpret_cast<const v2f*>(bptr + kn + nt * 16 * D_DIM);

            #pragma unroll
            for (int nt = 0; nt < 8; ++nt)
                c[nt] = __builtin_amdgcn_wmma_f32_16x16x4_f32(
                    false, a, false, b[nt], (short)0, c[nt], false, false);

            a = an;
            #pragma unroll
            for (int nt = 0; nt < 8; ++nt) b[nt] = bn[nt];
        }

        // Fold scores: dist-part = ||p||^2 - 2 * (x . p)
        #pragma unroll
        for (int nt = 0; nt < 8; ++nt) {
            int Nidx = nBase + nt * 16 + m;
            float pn = pnorm[Nidx];
            #pragma unroll
            for (int r = 0; r < 8; ++r) {
                float s = pn - 2.0f * c[nt][r];
                if (s < minv[r]) { minv[r] = s; mini[r] = Nidx; }  // strict < keeps lowest index
            }
        }
    }

    // Cross-lane min-reduce within 16-lane groups (lanes 0-15: rows 0-7 at
    // VGPR r; lanes 16-31: rows 8-15). XOR strides <= 8 never cross groups.
    #pragma unroll
    for (int off = 8; off >= 1; off >>= 1) {
        #pragma unroll
        for (int r = 0; r < 8; ++r) {
            float ov = __shfl_xor(minv[r], off);
            int   oi = __shfl_xor(mini[r], off);
            if (ov < minv[r] || (ov == minv[r] && oi < mini[r])) {
                minv[r] = ov; mini[r] = oi;
            }
        }
    }

    if (lane == 0) {
        #pragma unroll
        for (int r = 0; r < 8; ++r) sIdx[wave * 16 + r] = mini[r];
    }
    if (lane == 16) {
        #pragma unroll
        for (int r = 0; r < 8; ++r) sIdx[wave * 16 + 8 + r] = mini[r];
    }
    __syncthreads();

    // Gather + write proto, residual; accumulate squared error
    float lsum = 0.f;
    for (int row = 0; row < 16; ++row) {
        int j = sIdx[wave * 16 + row];
        size_t gro = (size_t)(rowBase + row) * D_DIM;
        const float4* xr = reinterpret_cast<const float4*>(x + gro);
        const float4* pr = reinterpret_cast<const float4*>(protos + (size_t)j * D_DIM);
        float4* po = reinterpret_cast<float4*>(proto_out + gro);
        float4* ro = reinterpret_cast<float4*>(resid_out + gro);
        #pragma unroll
        for (int t = 0; t < 2; ++t) {
            int i4 = lane + t * 32;           // 64 float4 per row, 32 lanes
            float4 xv = xr[i4];
            float4 pv = pr[i4];
            po[i4] = pv;
            float4 rv;
            rv.x = xv.x - pv.x; rv.y = xv.y - pv.y;
            rv.z = xv.z - pv.z; rv.w = xv.w - pv.w;
            ro[i4] = rv;
            lsum += rv.x * rv.x + rv.y * rv.y + rv.z * rv.z + rv.w * rv.w;
        }
    }

    // Wave-level sum reduce
    #pragma unroll
    for (int off = 16; off >= 1; off >>= 1) lsum += __shfl_xor(lsum, off);
    if (lane == 0) partials[waveGlobal] = lsum;
}

// ---------------------------------------------------------------------------
// Kernel 3: deterministic fixed-order reduction of per-wave partials
// loss = 1.25 * sum / numel
// ---------------------------------------------------------------------------
__global__ void loss_reduce(const float* __restrict__ partials, int n,
                            float* __restrict__ out, float scale) {
    __shared__ float sm[256];
    int tid = threadIdx.x;
    float s = 0.f;
    for (int i = tid; i < n; i += 256) s += partials[i];
    sm[tid] = s;
    __syncthreads();
    for (int off = 128; off > 0; off >>= 1) {
        if (tid < off) sm[tid] += sm[tid + off];
        __syncthreads();
    }
    if (tid == 0) out[0] = sm[0] * scale;
}

// ---------------------------------------------------------------------------
extern "C" void kernel_launch(void* const* d_in, const int* in_sizes, int n_in,
                              void* d_out, int out_size, void* d_ws, size_t ws_size,
                              hipStream_t stream) {
    const float* x      = (const float*)d_in[0];
    const float* protos = (const float*)d_in[1];

    const int xN = in_sizes[0];      // 64*128*8*256 = 16777216
    const int pN = in_sizes[1];      // 512*256       = 131072
    const int T  = xN / D_DIM;       // 65536 rows
    const int P  = pN / D_DIM;       // 512 prototypes
    const int nWaves = T / 16;       // 4096

    float* pnorm    = (float*)d_ws;
    float* partials = pnorm + P;

    float* out       = (float*)d_out;
    float* proto_out = out;
    float* resid_out = out + (size_t)xN;
    float* loss_out  = out + 2 * (size_t)xN;

    pnorm_kernel<<<(P + 255) / 256, 256, 0, stream>>>(protos, pnorm, P);
    vq_main<<<T / 128, 256, 0, stream>>>(x, protos, pnorm,
                                         proto_out, resid_out, partials, P);
    loss_reduce<<<1, 256, 0, stream>>>(partials, nWaves, loss_out,
                                       1.25f / (float)xN);
}